// Entropy_bottleneck_38190849196804
// MI455X (gfx1250) — compile-verified
//
#include <hip/hip_runtime.h>
#include <hip/hip_bf16.h>

#define NCH   192
#define HW    4096            // 64*64 elements per (b,c) slab
#define NB    16
#define WSTRIDE 48            // padded per-channel weight record (floats)

// ---------------------------------------------------------------------------
// CDNA5 async global->LDS copy (B128) + wait, with inline-asm fallbacks
// ---------------------------------------------------------------------------
typedef int async_v4i __attribute__((vector_size(16)));
typedef __attribute__((address_space(1))) async_v4i* gbl_v4i_ptr;
typedef __attribute__((address_space(3))) async_v4i* lds_v4i_ptr;

__device__ __forceinline__ void async_copy_b128(const float* gsrc, float* lds) {
#if __has_builtin(__builtin_amdgcn_global_load_async_to_lds_b128)
    __builtin_amdgcn_global_load_async_to_lds_b128(
        (gbl_v4i_ptr)gsrc, (lds_v4i_ptr)lds, 0, 0);
#else
    unsigned lds_off = (unsigned)(unsigned long long)(void*)lds;  // low 32 bits of flat LDS addr == LDS byte offset
    asm volatile("global_load_async_to_lds_b128 %0, %1, off"
                 :: "v"(lds_off), "v"(gsrc)
                 : "memory");
#endif
}

__device__ __forceinline__ void wait_async0() {
#if __has_builtin(__builtin_amdgcn_s_wait_asynccnt)
    __builtin_amdgcn_s_wait_asynccnt(0);
#else
    asm volatile("s_wait_asynccnt 0" ::: "memory");
#endif
}

// ---------------------------------------------------------------------------
// Fast transcendentals — force the gfx1250 hardware V_TANH_F32 either way
// ---------------------------------------------------------------------------
__device__ __forceinline__ float fast_tanh(float x) {
#if __has_builtin(__builtin_amdgcn_tanhf)
    return __builtin_amdgcn_tanhf(x);
#else
    float r;
    asm("v_tanh_f32 %0, %1" : "=v"(r) : "v"(x));
    return r;
#endif
}

__device__ __forceinline__ float fast_sigmoid(float x) {
    float e = __builtin_amdgcn_exp2f(-1.44269504088896340f * x);
    return __builtin_amdgcn_rcpf(1.0f + e);
}

// softplus(x) = ln(1+e^x) = max(x,0) + ln(1+e^{-|x|}), via v_exp_f32/v_log_f32
__device__ __forceinline__ float fast_softplus(float x) {
    float ax = fabsf(x);
    float e  = __builtin_amdgcn_exp2f(-1.44269504088896340f * ax);   // e^{-|x|}
    float l  = __builtin_amdgcn_logf(1.0f + e) * 0.69314718055994531f; // ln(1+e^{-|x|})
    return fmaxf(x, 0.0f) + l;
}

// ---------------------------------------------------------------------------
// Packed per-channel constants (43 used, padded to 48 floats)
// [0..2]=softplus(m0)  [3..5]=b0  [6..8]=tanh(f0)
// [9..17]=softplus(m1) [18..20]=b1 [21..23]=tanh(f1)
// [24..32]=softplus(m2)[33..35]=b2 [36..38]=tanh(f2)
// [39..41]=softplus(m3)[42]=b3
// ---------------------------------------------------------------------------
struct Wts {
    float sp0[3], bb0[3], t0[3];
    float sp1[9], bb1[3], t1[3];
    float sp2[9], bb2[3], t2[3];
    float sp3[3], bb3;
};

// ---------------------------------------------------------------------------
// The cumulative-logits chain for one scalar input
// ---------------------------------------------------------------------------
__device__ __forceinline__ float chain(float v, const Wts& w) {
    // layer 0: 1 -> 3, then tanh gate
    float a0 = fmaf(w.sp0[0], v, w.bb0[0]);
    float a1 = fmaf(w.sp0[1], v, w.bb0[1]);
    float a2 = fmaf(w.sp0[2], v, w.bb0[2]);
    a0 = fmaf(w.t0[0], fast_tanh(a0), a0);
    a1 = fmaf(w.t0[1], fast_tanh(a1), a1);
    a2 = fmaf(w.t0[2], fast_tanh(a2), a2);
    // layer 1: 3 -> 3, tanh gate
    float y0 = fmaf(w.sp1[0], a0, fmaf(w.sp1[1], a1, fmaf(w.sp1[2], a2, w.bb1[0])));
    float y1 = fmaf(w.sp1[3], a0, fmaf(w.sp1[4], a1, fmaf(w.sp1[5], a2, w.bb1[1])));
    float y2 = fmaf(w.sp1[6], a0, fmaf(w.sp1[7], a1, fmaf(w.sp1[8], a2, w.bb1[2])));
    y0 = fmaf(w.t1[0], fast_tanh(y0), y0);
    y1 = fmaf(w.t1[1], fast_tanh(y1), y1);
    y2 = fmaf(w.t1[2], fast_tanh(y2), y2);
    // layer 2: 3 -> 3, tanh gate
    float z0 = fmaf(w.sp2[0], y0, fmaf(w.sp2[1], y1, fmaf(w.sp2[2], y2, w.bb2[0])));
    float z1 = fmaf(w.sp2[3], y0, fmaf(w.sp2[4], y1, fmaf(w.sp2[5], y2, w.bb2[1])));
    float z2 = fmaf(w.sp2[6], y0, fmaf(w.sp2[7], y1, fmaf(w.sp2[8], y2, w.bb2[2])));
    z0 = fmaf(w.t2[0], fast_tanh(z0), z0);
    z1 = fmaf(w.t2[1], fast_tanh(z1), z1);
    z2 = fmaf(w.t2[2], fast_tanh(z2), z2);
    // layer 3: 3 -> 1
    return fmaf(w.sp3[0], z0, fmaf(w.sp3[1], z1, fmaf(w.sp3[2], z2, w.bb3)));
}

__device__ __forceinline__ float likelihood_of(float xv, const Wts& w) {
    float lo = chain(xv - 0.5f, w);
    float up = chain(xv + 0.5f, w);
    float sum = lo + up;
    float s = (sum > 0.0f) ? -1.0f : ((sum < 0.0f) ? 1.0f : 0.0f);
    float lk = fabsf(fast_sigmoid(s * up) - fast_sigmoid(s * lo));
    return fmaxf(lk, 1e-6f);
}

// ---------------------------------------------------------------------------
// Main kernel (defined FIRST so its asm appears in the disasm snippet):
// one block per (b,c) slab of 4096 contiguous floats. Async-copy the 16KB
// tile to LDS (4x B128 per lane), overlap with uniform SGPR weight loads,
// then compute 16 elements/thread.
// ---------------------------------------------------------------------------
__global__ __launch_bounds__(256) void eb_main(const float* __restrict__ x,
                                               const float* __restrict__ wbuf,
                                               float* __restrict__ out) {
    __shared__ float tile[HW];
    const int tid  = threadIdx.x;
    const int slab = blockIdx.x;            // b*NCH + c
    const int c    = slab % NCH;
    const size_t base = (size_t)slab * HW;
    const float* gx = x + base;

    // 1) kick off async global->LDS copies (each lane copies what it consumes)
    #pragma unroll
    for (int k = 0; k < 4; ++k) {
        int off = tid * 4 + k * 1024;
        async_copy_b128(gx + off, &tile[off]);
    }

    // 2) uniform (block-constant) weight loads -> SGPRs, overlapped with copy
    Wts w;
    {
        const float* p = wbuf + c * WSTRIDE;
        float* wf = (float*)&w;
        #pragma unroll
        for (int i = 0; i < 43; ++i) wf[i] = p[i];
    }

    // 3) wait for this wave's async copies; each lane reads its own data
    wait_async0();

    #pragma unroll
    for (int k = 0; k < 4; ++k) {
        int off = tid * 4 + k * 1024;
        float4 xv = *reinterpret_cast<const float4*>(&tile[off]);
        float4 r;
        r.x = likelihood_of(xv.x, w);
        r.y = likelihood_of(xv.y, w);
        r.z = likelihood_of(xv.z, w);
        r.w = likelihood_of(xv.w, w);
        *reinterpret_cast<float4*>(out + base + off) = r;
    }
}

// ---------------------------------------------------------------------------
// One-shot per-channel weight preparation (192 threads total)
// ---------------------------------------------------------------------------
__global__ __launch_bounds__(192) void eb_prep(
                        const float* __restrict__ m0, const float* __restrict__ b0,
                        const float* __restrict__ m1, const float* __restrict__ b1,
                        const float* __restrict__ m2, const float* __restrict__ b2,
                        const float* __restrict__ m3, const float* __restrict__ b3,
                        const float* __restrict__ f0, const float* __restrict__ f1,
                        const float* __restrict__ f2, float* __restrict__ wbuf) {
    int c = blockIdx.x * blockDim.x + threadIdx.x;
    if (c >= NCH) return;
    float* o = wbuf + c * WSTRIDE;
    #pragma unroll
    for (int j = 0; j < 3; ++j) {
        o[0 + j]  = fast_softplus(m0[c * 3 + j]);
        o[3 + j]  = b0[c * 3 + j];
        o[6 + j]  = fast_tanh(f0[c * 3 + j]);
        o[18 + j] = b1[c * 3 + j];
        o[21 + j] = fast_tanh(f1[c * 3 + j]);
        o[33 + j] = b2[c * 3 + j];
        o[36 + j] = fast_tanh(f2[c * 3 + j]);
        o[39 + j] = fast_softplus(m3[c * 3 + j]);
    }
    #pragma unroll
    for (int k = 0; k < 9; ++k) {
        o[9 + k]  = fast_softplus(m1[c * 9 + k]);
        o[24 + k] = fast_softplus(m2[c * 9 + k]);
    }
    o[42] = b3[c];
    o[43] = 0.0f; o[44] = 0.0f; o[45] = 0.0f; o[46] = 0.0f; o[47] = 0.0f;
}

// ---------------------------------------------------------------------------
// Inputs (setup_inputs dict order):
//  0:x  1:m0 2:b0 3:m1 4:b1 5:m2 6:b2 7:m3 8:b3 9:f0 10:f1 11:f2
// ---------------------------------------------------------------------------
extern "C" void kernel_launch(void* const* d_in, const int* in_sizes, int n_in,
                              void* d_out, int out_size, void* d_ws, size_t ws_size,
                              hipStream_t stream) {
    (void)in_sizes; (void)n_in; (void)out_size; (void)ws_size;
    const float* x  = (const float*)d_in[0];
    const float* m0 = (const float*)d_in[1];
    const float* b0 = (const float*)d_in[2];
    const float* m1 = (const float*)d_in[3];
    const float* b1 = (const float*)d_in[4];
    const float* m2 = (const float*)d_in[5];
    const float* b2 = (const float*)d_in[6];
    const float* m3 = (const float*)d_in[7];
    const float* b3 = (const float*)d_in[8];
    const float* f0 = (const float*)d_in[9];
    const float* f1 = (const float*)d_in[10];
    const float* f2 = (const float*)d_in[11];
    float* wbuf = (float*)d_ws;               // NCH * WSTRIDE * 4 = 36,864 bytes

    eb_prep<<<1, NCH, 0, stream>>>(m0, b0, m1, b1, m2, b2, m3, b3, f0, f1, f2, wbuf);
    eb_main<<<NB * NCH, 256, 0, stream>>>(x, wbuf, (float*)d_out);
}